// MyGCN_49641232007691
// MI455X (gfx1250) — compile-verified
//
#include <hip/hip_runtime.h>

#define NN 100000
#define NE 640000
#define F1 320
#define F2 160

typedef __attribute__((ext_vector_type(2))) float v2f;
typedef __attribute__((ext_vector_type(8))) float v8f;

// Order-preserving float <-> uint map so scatter-max can use native u32 atomics.
__device__ __forceinline__ unsigned f2o(float f) {
    unsigned u = __float_as_uint(f);
    return (u & 0x80000000u) ? ~u : (u | 0x80000000u);
}
__device__ __forceinline__ float o2f(unsigned u) {
    return (u & 0x80000000u) ? __uint_as_float(u & 0x7fffffffu)
                             : __uint_as_float(~u);
}

__global__ void k_init_deg(unsigned* __restrict__ deg) {
    int i = blockIdx.x * blockDim.x + threadIdx.x;
    if (i < NN) deg[i] = 1u;  // self loop
}

__global__ void k_count_deg(const int* __restrict__ dst, unsigned* __restrict__ deg) {
    int e = blockIdx.x * blockDim.x + threadIdx.x;
    if (e < NE) atomicAdd(&deg[dst[e]], 1u);
}

__global__ void k_dinv(const unsigned* __restrict__ deg, float* __restrict__ dinv) {
    int i = blockIdx.x * blockDim.x + threadIdx.x;
    if (i < NN) dinv[i] = rsqrtf((float)deg[i]);
}

// h1 = x @ W1 (K=5); also seed aggregation buffer with the self-loop message
// (norm = dinv[i]^2) in order-preserving encoding.
__global__ void k_gemm1_init(const float* __restrict__ x, const float* __restrict__ W1,
                             const float* __restrict__ dinv,
                             float* __restrict__ h1, unsigned* __restrict__ enc1) {
    int id = blockIdx.x * blockDim.x + threadIdx.x;
    if (id >= NN * (F1 / 4)) return;
    int i = id / (F1 / 4);
    int f0 = (id % (F1 / 4)) * 4;
    float xv[5];
#pragma unroll
    for (int k = 0; k < 5; ++k) xv[k] = x[i * 5 + k];
    float4 acc = make_float4(0.f, 0.f, 0.f, 0.f);
#pragma unroll
    for (int k = 0; k < 5; ++k) {
        float4 w = *(const float4*)(W1 + k * F1 + f0);
        acc.x += xv[k] * w.x; acc.y += xv[k] * w.y;
        acc.z += xv[k] * w.z; acc.w += xv[k] * w.w;
    }
    *(float4*)(h1 + i * F1 + f0) = acc;
    float d = dinv[i], dsq = d * d;
    uint4 e;
    e.x = f2o(dsq * acc.x); e.y = f2o(dsq * acc.y);
    e.z = f2o(dsq * acc.z); e.w = f2o(dsq * acc.w);
    *(uint4*)(enc1 + i * F1 + f0) = e;
}

// Per-edge scatter-max: F/4 consecutive threads cover one edge's feature row
// -> coalesced gather of h[src] and coalesced atomic row to enc[dst].
template <int F>
__global__ void k_scatter(const int* __restrict__ src, const int* __restrict__ dst,
                          const float* __restrict__ dinv, const float* __restrict__ h,
                          unsigned* __restrict__ enc) {
    const int C = F / 4;
    int id = blockIdx.x * blockDim.x + threadIdx.x;
    if (id >= NE * C) return;
    int e = id / C;
    int f0 = (id % C) * 4;
    int s = src[e], d = dst[e];
    float norm = dinv[s] * dinv[d];
    float4 v = *(const float4*)(h + s * F + f0);
    unsigned* p = enc + d * F + f0;
    atomicMax(p + 0, f2o(norm * v.x));
    atomicMax(p + 1, f2o(norm * v.y));
    atomicMax(p + 2, f2o(norm * v.z));
    atomicMax(p + 3, f2o(norm * v.w));
}

// decode + bias + ReLU -> input of layer 2
__global__ void k_final1(const unsigned* __restrict__ enc1, const float* __restrict__ b1,
                         float* __restrict__ h1b) {
    int id = blockIdx.x * blockDim.x + threadIdx.x;
    if (id >= NN * (F1 / 4)) return;
    int i = id / (F1 / 4);
    int f0 = (id % (F1 / 4)) * 4;
    uint4 u = *(const uint4*)(enc1 + i * F1 + f0);
    float4 b = *(const float4*)(b1 + f0);
    float4 v;
    v.x = fmaxf(o2f(u.x) + b.x, 0.f);
    v.y = fmaxf(o2f(u.y) + b.y, 0.f);
    v.z = fmaxf(o2f(u.z) + b.z, 0.f);
    v.w = fmaxf(o2f(u.w) + b.w, 0.f);
    *(float4*)(h1b + i * F1 + f0) = v;
}

// h2 = A(100000x320) @ W2(320x160) with V_WMMA_F32_16X16X4_F32.
// Each wave owns 16 rows x all 160 cols (10 accumulator tiles). W2 is staged
// in LDS in 64x160 K-chunks (40 KB), stored K-PAIRED:
//   w2s[((k/2)*F2 + col)*2 + (k&1)] = W2[k0+k][col]
// so each lane's B fragment {W2[k+kb][col], W2[k+kb+1][col]} (kb in {0,2},
// k+kb even) is one aligned 8-byte ds_load feeding v_wmma directly — no
// register shuffles. Also seeds layer-2 self-loop messages.
#define KT 64
__global__ void __launch_bounds__(256)
k_gemm2_wmma(const float* __restrict__ A, const float* __restrict__ W2,
             const float* __restrict__ dinv,
             float* __restrict__ h2, unsigned* __restrict__ enc2) {
    __shared__ float w2s[KT * F2];  // 40 KB, pair-swizzled
    const int tid = threadIdx.x;
    const int lane = tid & 31;
    const int wave = tid >> 5;
    const int base_m = (blockIdx.x * 8 + wave) * 16;
    const bool active = (base_m < NN);  // NN % 16 == 0 -> wave-uniform
    const int lh = lane & 15;
    const int hi = (lane >> 4) & 1;
    const int kb = hi * 2;  // A holds K=0,1 in lanes 0-15 and K=2,3 in lanes 16-31

    v8f acc[10] = {};

    for (int k0 = 0; k0 < F1; k0 += KT) {
        __syncthreads();
        for (int idx = tid; idx < KT * F2; idx += 256) {
            int kt = idx / F2;
            int n  = idx - kt * F2;
            w2s[((kt >> 1) * F2 + n) * 2 + (kt & 1)] = W2[k0 * F2 + idx];
        }
        __syncthreads();
        if (active) {
            const float* arow = A + (base_m + lh) * F1 + k0 + kb;
            const v2f* bp = (const v2f*)w2s;
            for (int k = 0; k < KT; k += 4) {
                float2 av = *(const float2*)(arow + k);
                v2f a; a.x = av.x; a.y = av.y;
                int prow = ((k + kb) >> 1) * F2 + lh;  // pair row base + lane col
#pragma unroll
                for (int nt = 0; nt < 10; ++nt) {
                    v2f b = bp[prow + nt * 16];
                    acc[nt] = __builtin_amdgcn_wmma_f32_16x16x4_f32(
                        false, a, false, b, (short)0, acc[nt], false, false);
                }
            }
        }
    }
    if (active) {
#pragma unroll
        for (int r = 0; r < 8; ++r) {
            int m = base_m + r + hi * 8;  // D: VGPR r -> M=r (lanes 0-15), M=r+8 (16-31)
            float dv = dinv[m], dsq = dv * dv;
#pragma unroll
            for (int nt = 0; nt < 10; ++nt) {
                int col = nt * 16 + lh;
                float v = acc[nt][r];
                h2[m * F2 + col] = v;
                enc2[m * F2 + col] = f2o(dsq * v);
            }
        }
    }
}

// Fused: decode layer-2 max, add b2, multiply by Wl(160x2), add bl.
// One wave per node; shuffle reduction.
__global__ void k_out(const unsigned* __restrict__ enc2, const float* __restrict__ b2,
                      const float* __restrict__ Wl, const float* __restrict__ bl,
                      float* __restrict__ out) {
    int gwave = (blockIdx.x * blockDim.x + threadIdx.x) >> 5;
    int lane = threadIdx.x & 31;
    if (gwave >= NN) return;
    const int i = gwave;
    float a0 = 0.f, a1 = 0.f;
    {
        int f0 = lane * 4;  // f 0..127
        uint4 u = *(const uint4*)(enc2 + i * F2 + f0);
        float4 b = *(const float4*)(b2 + f0);
        float v0 = o2f(u.x) + b.x, v1 = o2f(u.y) + b.y;
        float v2 = o2f(u.z) + b.z, v3 = o2f(u.w) + b.w;
        a0 += v0 * Wl[(f0 + 0) * 2] + v1 * Wl[(f0 + 1) * 2] +
              v2 * Wl[(f0 + 2) * 2] + v3 * Wl[(f0 + 3) * 2];
        a1 += v0 * Wl[(f0 + 0) * 2 + 1] + v1 * Wl[(f0 + 1) * 2 + 1] +
              v2 * Wl[(f0 + 2) * 2 + 1] + v3 * Wl[(f0 + 3) * 2 + 1];
    }
    if (lane < 8) {
        int f0 = 128 + lane * 4;  // f 128..159
        uint4 u = *(const uint4*)(enc2 + i * F2 + f0);
        float4 b = *(const float4*)(b2 + f0);
        float v0 = o2f(u.x) + b.x, v1 = o2f(u.y) + b.y;
        float v2 = o2f(u.z) + b.z, v3 = o2f(u.w) + b.w;
        a0 += v0 * Wl[(f0 + 0) * 2] + v1 * Wl[(f0 + 1) * 2] +
              v2 * Wl[(f0 + 2) * 2] + v3 * Wl[(f0 + 3) * 2];
        a1 += v0 * Wl[(f0 + 0) * 2 + 1] + v1 * Wl[(f0 + 1) * 2 + 1] +
              v2 * Wl[(f0 + 2) * 2 + 1] + v3 * Wl[(f0 + 3) * 2 + 1];
    }
#pragma unroll
    for (int off = 16; off > 0; off >>= 1) {
        a0 += __shfl_down(a0, off, 32);
        a1 += __shfl_down(a1, off, 32);
    }
    if (lane == 0) {
        out[i * 2 + 0] = a0 + bl[0];
        out[i * 2 + 1] = a1 + bl[1];
    }
}

extern "C" void kernel_launch(void* const* d_in, const int* in_sizes, int n_in,
                              void* d_out, int out_size, void* d_ws, size_t ws_size,
                              hipStream_t stream) {
    const float* x  = (const float*)d_in[0];
    const int*   ei = (const int*)d_in[1];
    const int*   src = ei;
    const int*   dst = ei + NE;
    const float* W1 = (const float*)d_in[2];
    const float* b1 = (const float*)d_in[3];
    const float* W2 = (const float*)d_in[4];
    const float* b2 = (const float*)d_in[5];
    const float* Wl = (const float*)d_in[6];
    const float* bl = (const float*)d_in[7];

    // Workspace layout (all offsets keep 16B alignment):
    float*    wsf  = (float*)d_ws;
    float*    dinv = wsf;                       // NN floats
    unsigned* deg  = (unsigned*)(wsf + NN);     // NN u32
    float*    bufA = wsf + 2 * NN;              // NN*F1 floats: h1, then relu'd h1
    float*    bufB = bufA + (size_t)NN * F1;    // NN*F1 words: enc1, then h2+enc2
    unsigned* enc1 = (unsigned*)bufB;
    float*    h2   = bufB;                      // reuse after enc1 consumed
    unsigned* enc2 = (unsigned*)(bufB + (size_t)NN * F2);

    const int B = 256;
    k_init_deg <<<(NN + B - 1) / B, B, 0, stream>>>(deg);
    k_count_deg<<<(NE + B - 1) / B, B, 0, stream>>>(dst, deg);
    k_dinv     <<<(NN + B - 1) / B, B, 0, stream>>>(deg, dinv);

    k_gemm1_init<<<(NN * (F1 / 4) + B - 1) / B, B, 0, stream>>>(x, W1, dinv, bufA, enc1);
    k_scatter<F1><<<(NE * (F1 / 4) + B - 1) / B, B, 0, stream>>>(src, dst, dinv, bufA, enc1);
    k_final1<<<(NN * (F1 / 4) + B - 1) / B, B, 0, stream>>>(enc1, b1, bufA);

    k_gemm2_wmma<<<(NN / 16 + 7) / 8, 256, 0, stream>>>(bufA, W2, dinv, h2, enc2);
    k_scatter<F2><<<(NE * (F2 / 4) + B - 1) / B, B, 0, stream>>>(src, dst, dinv, h2, enc2);

    k_out<<<(NN * 32 + B - 1) / B, B, 0, stream>>>(enc2, b2, Wl, bl, (float*)d_out);
}